// Shift_gcn_local_23261542875773
// MI455X (gfx1250) — compile-verified
//
#include <hip/hip_runtime.h>

typedef __attribute__((ext_vector_type(2))) float v2f;
typedef __attribute__((ext_vector_type(8))) float v8f;

#define NODES 115
#define C_IN  64
#define D_OUT 64
#define T_DIM 128
#define N_BATCH 32
#define NT    (N_BATCH * T_DIM)       // 4096
#define COLS  (NODES * D_OUT)         // 7360
#define LDA   65                      // padded LDS stride (bank-conflict free)
#define BN_EPS 1e-5f

// ---------------------------------------------------------------------------
// Kernel 1: fused shift_in-gather + per-node GEMM (fp32 WMMA) + bias
//           + per-column sum / sum-of-squares for BN (atomics into scratch)
// grid = (256 M-tiles, 115 nodes), block = 128 threads (4 waves, one 16-wide
// d-tile per wave).
// ---------------------------------------------------------------------------
__global__ __launch_bounds__(128)
void k1_gemm(const float* __restrict__ x0, const float* __restrict__ W,
             const float* __restrict__ bias, const int* __restrict__ shift_in,
             float* __restrict__ Y, float* __restrict__ colsum,
             float* __restrict__ colsq)
{
    __shared__ float Alds[16 * LDA];

    const int mtile  = blockIdx.x;          // 0..255  (16 nt-rows each)
    const int w      = blockIdx.y;          // node 0..114
    const int ntbase = mtile * 16;
    const int tid    = threadIdx.x;

    // Cooperative gather of the 16x64 A tile (shift_in applied on the fly).
    // A[row][cc] = x_perm[ntbase+row, shift_in[w*64+cc]]
    //            = x0[n, s%64, t, s/64]  with s = shift_in[w*64+cc]
    #pragma unroll
    for (int i = 0; i < 8; ++i) {
        int idx  = tid + i * 128;           // 0..1023
        int row  = idx >> 6;
        int col  = idx & 63;
        int s    = shift_in[w * C_IN + col];
        int vsrc = s >> 6;                  // s / 64
        int csrc = s & 63;                  // s % 64
        int nt   = ntbase + row;
        int n    = nt >> 7;
        int t    = nt & 127;
        Alds[row * LDA + col] =
            x0[((size_t)(n * C_IN + csrc) * T_DIM + t) * NODES + vsrc];
    }
    __syncthreads();

    const int wave = tid >> 5;              // d-tile 0..3
    const int lane = tid & 31;
    const int d0   = wave * 16;
    const int mrow = lane & 15;             // A: M = lane % 16
    const int koff = (lane >> 4) * 2;       // A/B: lanes 16-31 hold K=2,3
    const int ncol = d0 + (lane & 15);      // B/C: N = lane % 16

    // K = 64 via 16 chained V_WMMA_F32_16X16X4_F32 (full fp32 precision)
    v8f acc = {};
    #pragma unroll
    for (int k0 = 0; k0 < C_IN; k0 += 4) {
        v2f a, b;
        a.x = Alds[mrow * LDA + k0 + koff];
        a.y = Alds[mrow * LDA + k0 + koff + 1];
        b.x = W[(k0 + koff)     * D_OUT + ncol];
        b.y = W[(k0 + koff + 1) * D_OUT + ncol];
        acc = __builtin_amdgcn_wmma_f32_16x16x4_f32(
                  false, a, false, b, (short)0, acc, false, false);
    }

    // Epilogue: bias, store Y, per-column BN partial stats.
    const float bv  = bias[ncol];
    const int   col = w * D_OUT + ncol;     // column of Y (0..7359)
    float s = 0.f, q = 0.f;
    #pragma unroll
    for (int r = 0; r < 8; ++r) {
        int   M   = r + (lane >> 4) * 8;    // C/D layout: lanes 16-31 -> M+8
        float val = acc[r] + bv;
        Y[(size_t)(ntbase + M) * COLS + col] = val;
        s += val;
        q += val * val;
    }
    atomicAdd(&colsum[col], s);
    atomicAdd(&colsq[col], q);
}

// ---------------------------------------------------------------------------
// Kernel 2: finalize BN statistics (7360 columns)
// ---------------------------------------------------------------------------
__global__ void k2_stats(const float* __restrict__ colsum,
                         const float* __restrict__ colsq,
                         float* __restrict__ mean, float* __restrict__ invstd)
{
    int j = blockIdx.x * blockDim.x + threadIdx.x;
    if (j < COLS) {
        float m = colsum[j] * (1.0f / NT);
        float v = colsq[j] * (1.0f / NT) - m * m;
        mean[j]   = m;
        invstd[j] = rsqrtf(v + BN_EPS);
    }
}

// ---------------------------------------------------------------------------
// Kernel 3: shift_out-gather + BN apply + residual + ReLU, written directly
// in x0's (n, d, t, v) layout (fully coalesced output; Y gathers hit L2).
// ---------------------------------------------------------------------------
__global__ __launch_bounds__(256)
void k3_out(const float* __restrict__ x0, const float* __restrict__ Y,
            const float* __restrict__ mean, const float* __restrict__ invstd,
            const float* __restrict__ gamma, const float* __restrict__ beta,
            const int* __restrict__ shift_out, float* __restrict__ out)
{
    const size_t total = (size_t)N_BATCH * D_OUT * T_DIM * NODES;
    size_t id = (size_t)blockIdx.x * blockDim.x + threadIdx.x;
    if (id >= total) return;

    int    vv = (int)(id % NODES);
    size_t r  = id / NODES;
    int    t  = (int)(r % T_DIM);  r /= T_DIM;
    int    dc = (int)(r % D_OUT);
    int    n  = (int)(r / D_OUT);

    int j  = vv * D_OUT + dc;               // column of shifted matrix
    int so = shift_out[j];                  // -> column of Y
    int nt = n * T_DIM + t;

    float y   = Y[(size_t)nt * COLS + so];
    float xb  = (y - mean[so]) * invstd[so] * gamma[j] + beta[j];
    float res = xb + x0[id];
    out[id] = res > 0.f ? res : 0.f;
}

// ---------------------------------------------------------------------------
extern "C" void kernel_launch(void* const* d_in, const int* in_sizes, int n_in,
                              void* d_out, int out_size, void* d_ws, size_t ws_size,
                              hipStream_t stream)
{
    const float* x0        = (const float*)d_in[0];
    const float* W         = (const float*)d_in[1];
    const float* bias      = (const float*)d_in[2];
    const float* gamma     = (const float*)d_in[3];
    const float* beta      = (const float*)d_in[4];
    const int*   shift_in  = (const int*)d_in[5];
    const int*   shift_out = (const int*)d_in[6];

    // Workspace layout: Y (4096*7360 f32) | colsum | colsq | mean | invstd
    float* Y      = (float*)d_ws;
    float* colsum = Y + (size_t)NT * COLS;
    float* colsq  = colsum + COLS;
    float* mean   = colsq + COLS;
    float* invstd = mean + COLS;

    hipMemsetAsync(colsum, 0, 2 * COLS * sizeof(float), stream);

    dim3 g1(NT / 16, NODES);                // 256 x 115 blocks
    k1_gemm<<<g1, 128, 0, stream>>>(x0, W, bias, shift_in, Y, colsum, colsq);

    k2_stats<<<(COLS + 255) / 256, 256, 0, stream>>>(colsum, colsq, mean, invstd);

    const size_t total = (size_t)N_BATCH * D_OUT * T_DIM * NODES;
    k3_out<<<(unsigned)((total + 255) / 256), 256, 0, stream>>>(
        x0, Y, mean, invstd, gamma, beta, shift_out, (float*)d_out);
}